// JointsMSELossCoco_67800353734961
// MI455X (gfx1250) — compile-verified
//
#include <hip/hip_runtime.h>
#include <math.h>

typedef float v2f __attribute__((ext_vector_type(2)));
typedef float v8f __attribute__((ext_vector_type(8)));

#define B_    128
#define J_    17
#define H_    112
#define HW_   (H_ * H_)
#define BJ_   (B_ * J_)
#define RAD_  4

// d_out layout (floats): [loss(1)][x_pred(BJ*2)][tt(BJ*HW)][tw_new(BJ)]
#define XPRED_OFF 1
#define TT_OFF    (1 + BJ_ * 2)
#define TW_OFF    (TT_OFF + BJ_ * HW_)

__device__ __forceinline__ int imin_(int a, int b) { return a < b ? a : b; }
__device__ __forceinline__ int imax_(int a, int b) { return a > b ? a : b; }

// ws[0] = sum((h-t)^2), ws[1] = sum(diff2^2), ws[2] = sum(vis2)
__global__ void init_ws_kernel(float* ws) {
    if (threadIdx.x < 4) ws[threadIdx.x] = 0.0f;
}

// One block per (b,j): fused argmax + MSE partial + x_pred + tw_new + d2 partial.
// Streams heatmap/target once with b128 loads.
__global__ void argmax_mse_kernel(const float* __restrict__ heatmap,
                                  const float* __restrict__ target,
                                  const float* __restrict__ offset,
                                  const float* __restrict__ joints,
                                  const float* __restrict__ joints_vis,
                                  const float* __restrict__ tw,
                                  float* __restrict__ out,
                                  float* __restrict__ ws) {
    __shared__ float sval[256];
    __shared__ int   sidx[256];
    const int bj  = blockIdx.x;
    const int b   = bj / J_;
    const int j   = bj % J_;
    const int tid = threadIdx.x;

    const float4* hm4 = (const float4*)(heatmap + (size_t)bj * HW_);
    const float4* tg4 = (const float4*)(target  + (size_t)bj * HW_);

    float best = -INFINITY;
    int   bidx = HW_;
    float sse  = 0.0f;
    for (int i = tid; i < HW_ / 4; i += 256) {
        const float4 h = hm4[i];
        const float4 t = tg4[i];
        const float d0 = h.x - t.x, d1 = h.y - t.y, d2 = h.z - t.z, d3 = h.w - t.w;
        sse += d0 * d0 + d1 * d1 + d2 * d2 + d3 * d3;
        const int base = i * 4;
        if (h.x > best) { best = h.x; bidx = base + 0; }
        if (h.y > best) { best = h.y; bidx = base + 1; }
        if (h.z > best) { best = h.z; bidx = base + 2; }
        if (h.w > best) { best = h.w; bidx = base + 3; }
    }
    sval[tid] = best; sidx[tid] = bidx;
    __syncthreads();
    for (int s = 128; s > 0; s >>= 1) {
        if (tid < s) {
            float v2 = sval[tid + s]; int i2 = sidx[tid + s];
            if (v2 > sval[tid] || (v2 == sval[tid] && i2 < sidx[tid])) {
                sval[tid] = v2; sidx[tid] = i2;
            }
        }
        __syncthreads();
    }
    const int am = sidx[0];
    __syncthreads();

    // second reduction: SSE (reuse sval)
    sval[tid] = sse;
    __syncthreads();
    for (int s = 128; s > 0; s >>= 1) {
        if (tid < s) sval[tid] += sval[tid + s];
        __syncthreads();
    }

    if (tid == 0) {
        atomicAdd(&ws[0], sval[0]);

        const int yc = am / H_, xc = am % H_;
        const size_t obase = (size_t)b * (2 * J_) * HW_;
        const float offx = offset[obase + (size_t)j        * HW_ + am];
        const float offy = offset[obase + (size_t)(J_ + j) * HW_ + am];
        const float xp0 = (offx + (float)xc) * 2.0f;   // SCALE = 2
        const float xp1 = (offy + (float)yc) * 2.0f;
        out[XPRED_OFF + bj * 2 + 0] = xp0;
        out[XPRED_OFF + bj * 2 + 1] = xp1;

        const float jx = joints[bj * 3 + 0], jy = joints[bj * 3 + 1];
        const float vx = joints_vis[bj * 3 + 0], vy = joints_vis[bj * 3 + 1];
        const float w  = tw[bj];
        const int xi = (int)truncf(jx * 0.5f);
        const int yi = (int)truncf(jy * 0.5f);
        const bool inb    = (xi >= 0) && (xi <= H_ - 1) && (yi >= 0) && (yi <= H_ - 1);
        const bool active = truncf(w) >= 0.5f;
        out[TW_OFF + bj] = (active && !inb) ? 0.0f : w;

        const float dx = (xp0 - jx) * vx * (1.0f / 112.0f);
        const float dy = (xp1 - jy) * vy * (1.0f / 112.0f);
        atomicAdd(&ws[1], dx * dx + dy * dy);
        atomicAdd(&ws[2], vx + vy);
    }
}

// One block (256 thr = 8 waves) per (b,j): rank-1 Gaussian target via
// WMMA f32 16x16x4 outer products. Wave w (w<7) owns tile row ty=w.
__global__ void tt_wmma_kernel(const float* __restrict__ joints,
                               const float* __restrict__ tw,
                               float* __restrict__ out) {
    __shared__ float sy[H_];
    __shared__ float sx[H_];
    const int bj  = blockIdx.x;
    const int tid = threadIdx.x;

    const float jx = joints[bj * 3 + 0], jy = joints[bj * 3 + 1];
    const float w  = tw[bj];
    const int xi = (int)truncf(jx * 0.5f);
    const int yi = (int)truncf(jy * 0.5f);
    const bool inb    = (xi >= 0) && (xi <= H_ - 1) && (yi >= 0) && (yi <= H_ - 1);
    const bool active = truncf(w) >= 0.5f;
    const bool valid  = active && inb;
    const int xic = imin_(imax_(xi, 0), H_ - 1);
    const int yic = imin_(imax_(yi, 0), H_ - 1);

    // normalized 9-tap gaussian
    float kk[2 * RAD_ + 1];
    float ksum = 0.0f;
    #pragma unroll
    for (int t = 0; t < 2 * RAD_ + 1; ++t) {
        float x = (float)(t - RAD_);
        kk[t] = expf(-0.5f * x * x);
        ksum += kk[t];
    }
    const float kinv = 1.0f / ksum;
    #pragma unroll
    for (int t = 0; t < 2 * RAD_ + 1; ++t) kk[t] *= kinv;

    // 1-D blurred deltas with symmetric reflection
    for (int i = tid; i < 2 * H_; i += 256) {
        const int pos = (i < H_) ? i : (i - H_);
        const int ctr = (i < H_) ? yic : xic;
        float acc = 0.0f;
        #pragma unroll
        for (int t = 0; t < 2 * RAD_ + 1; ++t) {
            int p = pos + t - RAD_;
            p = (p < 0) ? (-p - 1) : p;
            p = (p >= H_) ? (2 * H_ - 1 - p) : p;
            acc += (p == ctr) ? kk[t] : 0.0f;
        }
        if (i < H_) sy[pos] = acc; else sx[pos] = acc;
    }
    __syncthreads();

    // min/max of rank-1 map = products of 1-D extrema (all values >= 0)
    float miny = INFINITY, maxy = -INFINITY, minx = INFINITY, maxx = -INFINITY;
    for (int i = 0; i < H_; ++i) {
        float a = sy[i]; miny = fminf(miny, a); maxy = fmaxf(maxy, a);
        float c = sx[i]; minx = fminf(minx, c); maxx = fmaxf(maxx, c);
    }
    const float gmin = miny * minx, gmax = maxy * maxx;
    const float inv    = 1.0f / (gmax - gmin);
    const float ascale = valid ? inv : 0.0f;
    const float bias   = valid ? (-gmin * inv) : 0.0f;

    const int lane  = tid & 31;
    const int l15   = lane & 15;
    const float lm  = (lane < 16) ? 1.0f : 0.0f;  // branchless K=0 lane mask
    // wave-uniform tile row: waves 0..6 own stripes, wave 7 idles
    const int ty = __builtin_amdgcn_readfirstlane(tid >> 5);
    if (ty < 7) {
        float* base = out + TT_OFF + (size_t)bj * HW_
                          + (size_t)(ty * 16 + ((lane >= 16) ? 8 : 0)) * H_;
        // A 16x4: K=0 column = gy*ascale in lanes 0-15; all other K zero
        v2f A;  A.x = sy[ty * 16 + l15] * ascale * lm;  A.y = 0.0f;
        v8f C;
        #pragma unroll
        for (int r = 0; r < 8; ++r) C[r] = bias;
        for (int tx = 0; tx < 7; ++tx) {   // uniform scalar loop, EXEC stays all-1s
            // B 4x16: row K=0 = gx in lanes 0-15 of first VGPR; other rows zero
            v2f Bm; Bm.x = sx[tx * 16 + l15] * lm; Bm.y = 0.0f;
            v8f D = __builtin_amdgcn_wmma_f32_16x16x4_f32(
                /*neg_a=*/false, A, /*neg_b=*/false, Bm,
                /*c_mod=*/(short)0, C, /*reuse_a=*/false, /*reuse_b=*/false);
            // D: VGPR r -> M = r (+8 for lanes 16-31), N = lane & 15
            const int col = tx * 16 + l15;
            #pragma unroll
            for (int r = 0; r < 8; ++r)
                base[(size_t)r * H_ + col] = D[r];
        }
    }
}

__global__ void finalize_kernel(const float* __restrict__ ws, float* __restrict__ out) {
    if (threadIdx.x == 0 && blockIdx.x == 0) {
        const float d1 = ws[0] / (float)BJ_;
        const float n2 = ws[2] * 0.5f;
        const float d2 = 0.5f * sqrtf(ws[1]) / n2;
        out[0] = d1 + d2;
    }
}

extern "C" void kernel_launch(void* const* d_in, const int* in_sizes, int n_in,
                              void* d_out, int out_size, void* d_ws, size_t ws_size,
                              hipStream_t stream) {
    const float* offset     = (const float*)d_in[0];
    const float* heatmap    = (const float*)d_in[1];
    const float* target     = (const float*)d_in[2];
    const float* tweight    = (const float*)d_in[3];
    const float* joints     = (const float*)d_in[4];
    const float* joints_vis = (const float*)d_in[5];
    float* out = (float*)d_out;
    float* ws  = (float*)d_ws;

    init_ws_kernel<<<1, 64, 0, stream>>>(ws);
    argmax_mse_kernel<<<BJ_, 256, 0, stream>>>(heatmap, target, offset, joints,
                                               joints_vis, tweight, out, ws);
    tt_wmma_kernel<<<BJ_, 256, 0, stream>>>(joints, tweight, out);
    finalize_kernel<<<1, 1, 0, stream>>>(ws, out);
}